// TurboQuantValue_59588376265059
// MI455X (gfx1250) — compile-verified
//
#include <hip/hip_runtime.h>

typedef float v2f __attribute__((ext_vector_type(2)));
typedef float v4f __attribute__((ext_vector_type(4)));
typedef float v8f __attribute__((ext_vector_type(8)));

#define TQ_DIM            128
#define TQ_STRIDE         132      // padded LDS row stride (words) to spread banks
#define TQ_ROWS_PER_WAVE  8
#define TQ_ROWS_PER_BLOCK 64       // 8 waves * 8 rows
#define TQ_RSQRT_DIM      0.08838834764831845f   // 1/sqrt(128)
#define TQ_SQRT_DIM       11.313708498984761f    // sqrt(128)
#define TQ_LAPC           0.456f
#define TQ_EPS            1e-8f

// Async global<->LDS copies (CDNA5 ASYNCcnt path), with fallback to VGPR staging.
#if defined(__gfx1250__) &&                                                    \
    __has_builtin(__builtin_amdgcn_global_load_async_to_lds_b128) &&           \
    __has_builtin(__builtin_amdgcn_global_store_async_from_lds_b128)
#define TQ_ASYNC 1
#else
#define TQ_ASYNC 0
#endif

#if TQ_ASYNC
// Builtin parameter pointee type is int4 (per hipcc diagnostic):
//   '__attribute__((__vector_size__(4 * sizeof(int)))) int __device__ *'
typedef int tq_v4i __attribute__((__vector_size__(4 * sizeof(int))));
typedef __attribute__((address_space(1))) tq_v4i tq_g_v4i;   // global side
typedef __attribute__((address_space(3))) tq_v4i tq_l_v4i;   // LDS side
#endif

// D += A*B over K=16 as four K=4 f32 WMMAs (A = constant +-1 matrix, B = data)
__device__ __forceinline__ v8f tq_wmma16(const v2f* A, const v2f* B) {
  v8f c = {0.f, 0.f, 0.f, 0.f, 0.f, 0.f, 0.f, 0.f};
  c = __builtin_amdgcn_wmma_f32_16x16x4_f32(false, A[0], false, B[0], (short)0, c, false, false);
  c = __builtin_amdgcn_wmma_f32_16x16x4_f32(false, A[1], false, B[1], (short)0, c, false, false);
  c = __builtin_amdgcn_wmma_f32_16x16x4_f32(false, A[2], false, B[2], (short)0, c, false, false);
  c = __builtin_amdgcn_wmma_f32_16x16x4_f32(false, A[3], false, B[3], (short)0, c, false, false);
  return c;
}

// In-place H128 = (H16 (x) H8) on this wave's 8 rows in smem (natural layout:
// row r at smem[wb + r*TQ_STRIDE + 0..127]).
//   bmul: per-lane multiplier applied to stage-1 B operands (input sign fold)
//   dmul: per-lane multiplier applied to stage-2 D store (1/sqrt(128) and
//         optionally the output sign fold)
// Rows are wave-private; same-wave LDS ordering makes this barrier-free.
__device__ __forceinline__ void tq_hadamard128(float* __restrict__ smem, int wb, int L,
                                               const v2f* A1, const v2f* A2,
                                               const v2f* bmul, const float* dmul) {
  const int n  = L & 15;
  const int hi = L >> 4;

  // ---- stage 1: U = H16 * (bmul .* Z) ; tile tj covers rows 2tj, 2tj+1 ----
  {
    const int baseB = wb + (n >> 3) * TQ_STRIDE + (n & 7) + 16 * hi;
    const int baseD = wb + (n >> 3) * TQ_STRIDE + (n & 7) + 64 * hi;
#pragma unroll
    for (int tj = 0; tj < 4; ++tj) {
      const int tb = 2 * tj * TQ_STRIDE;
      v2f B[4];
#pragma unroll
      for (int k = 0; k < 4; ++k) {
        B[k][0] = smem[baseB + tb + 32 * k + 0] * bmul[k][0];
        B[k][1] = smem[baseB + tb + 32 * k + 8] * bmul[k][1];
      }
      v8f c = tq_wmma16(A1, B);
#pragma unroll
      for (int v = 0; v < 8; ++v) smem[baseD + tb + 8 * v] = c[v];
    }
  }

  // ---- stage 2: Y = blkdiag(H8,H8) * U^T-per-row ; scaled store ----
  {
    const int baseB = wb + 8 * n + 2 * hi;
    const int baseD = wb + hi * TQ_STRIDE + 8 * n;
#pragma unroll
    for (int tj = 0; tj < 4; ++tj) {
      const int tb = 2 * tj * TQ_STRIDE;
      v2f B[4];
#pragma unroll
      for (int k = 0; k < 4; ++k) {
        const int a = baseB + tb + (k >> 1) * TQ_STRIDE + 4 * (k & 1);
        B[k][0] = smem[a + 0];
        B[k][1] = smem[a + 1];
      }
      v8f c = tq_wmma16(A2, B);
#pragma unroll
      for (int v = 0; v < 8; ++v) smem[baseD + tb + v] = c[v] * dmul[v];
    }
  }
}

__global__ __launch_bounds__(256) void tq_kernel(const float* __restrict__ x,
                                                 const float* __restrict__ signs,
                                                 float* __restrict__ out,
                                                 long long totalRows) {
  __shared__ float smem[TQ_ROWS_PER_BLOCK * TQ_STRIDE];

  const int tid  = threadIdx.x;
  const int wave = tid >> 5;
  const int L    = tid & 31;
  const int wb   = wave * TQ_ROWS_PER_WAVE * TQ_STRIDE;

  const long long rowBase =
      (long long)blockIdx.x * TQ_ROWS_PER_BLOCK + (long long)wave * TQ_ROWS_PER_WAVE;
  if (rowBase + TQ_ROWS_PER_WAVE > totalRows) return;  // wave-uniform guard

  const float4* gin  = reinterpret_cast<const float4*>(x   + rowBase * TQ_DIM);
  float4*       gout = reinterpret_cast<float4*>      (out + rowBase * TQ_DIM);

  // ---- load: global -> LDS (pure copy; signs are folded into the transforms)
#if TQ_ASYNC
#pragma unroll
  for (int i = 0; i < 8; ++i) {
    const float* gp = reinterpret_cast<const float*>(gin + i * 32 + L);
    float*       lp = &smem[wb + i * TQ_STRIDE + 4 * L];
    __builtin_amdgcn_global_load_async_to_lds_b128(
        (tq_g_v4i*)(tq_v4i*)gp, (tq_l_v4i*)(tq_v4i*)lp, 0, 0);
  }
#else
#pragma unroll
  for (int i = 0; i < 8; ++i) {
    const float4 v = gin[i * 32 + L];
    v4f w;
    w[0] = v.x; w[1] = v.y; w[2] = v.z; w[3] = v.w;
    *reinterpret_cast<v4f*>(&smem[wb + i * TQ_STRIDE + 4 * L]) = w;
  }
#endif

  // ---- constant A-operands (computed once, reused by every tile) ----
  // A layout (16x4 f32): vgpr j, lane L -> element (m = L&15, k = 2*(L>>4)+j)
  v2f A1[4], A2[4];
  {
    const int i16 = L & 15;
    const int hi  = L >> 4;
#pragma unroll
    for (int k = 0; k < 4; ++k) {
#pragma unroll
      for (int j = 0; j < 2; ++j) {
        const int col = 4 * k + 2 * hi + j;
        A1[k][j] = (__popc(i16 & col) & 1) ? -1.f : 1.f;                  // H16
        A2[k][j] = ((i16 >> 3) == (col >> 3))                              // blkdiag(H8,H8)
                       ? ((__popc(i16 & col & 7) & 1) ? -1.f : 1.f)
                       : 0.f;
      }
    }
  }

  // ---- sign folds (per-lane, tile-invariant) ----
  // forward stage-1 B element (m,n) is x_row[8m+(n&7)], m = 4k+2*hi+j
  // inverse stage-2 D element (m,n,v) lands at output position 8n+v
  v2f sgnB1[4], onesB[4];
  float dmulF[8], dmulI[8];
  {
    const int hi = L >> 4;
#pragma unroll
    for (int k = 0; k < 4; ++k) {
#pragma unroll
      for (int j = 0; j < 2; ++j) {
        const int idx = 32 * k + 16 * hi + 8 * j + (L & 7);
        sgnB1[k][j] = signs[idx] >= 0.5f ? 1.f : -1.f;
        onesB[k][j] = 1.f;
      }
    }
#pragma unroll
    for (int v = 0; v < 8; ++v) {
      dmulF[v] = TQ_RSQRT_DIM;
      const int idx = 8 * (L & 15) + v;
      dmulI[v] = (signs[idx] >= 0.5f ? 1.f : -1.f) * TQ_RSQRT_DIM;
    }
  }

#if TQ_ASYNC
  asm volatile("s_wait_asynccnt 0" ::: "memory");  // LDS tile ready for DS reads
#endif

  // ---- forward transform: r = (1/sqrt(128)) * H128 * (x*s) ----
  tq_hadamard128(smem, wb, L, A1, A2, sgnB1, dmulF);

  // ---- per-row reductions: 4 lanes per row ----
  const int rrow  = L >> 2;
  const int rq    = L & 3;
  const int rbase = wb + rrow * TQ_STRIDE + rq * 32;
  float sumsq = 0.f, sumabs = 0.f;
#pragma unroll
  for (int t = 0; t < 8; ++t) {
    const v4f v = *reinterpret_cast<const v4f*>(&smem[rbase + 4 * t]);
#pragma unroll
    for (int e = 0; e < 4; ++e) {
      sumsq  += v[e] * v[e];
      sumabs += fabsf(v[e]);
    }
  }
  sumsq  += __shfl_xor(sumsq, 1, 32);
  sumsq  += __shfl_xor(sumsq, 2, 32);
  sumabs += __shfl_xor(sumabs, 1, 32);
  sumabs += __shfl_xor(sumabs, 2, 32);

  const float norm  = sqrtf(sumsq);
  const float S     = TQ_SQRT_DIM / (norm + TQ_EPS);       // u = r * S
  const float scale = TQ_LAPC * (S * sumabs * (1.0f / 128.0f));
  const float qmul  = S / (scale + TQ_EPS);                // q = round(r*qmul)
  const float deq   = scale * (norm * TQ_RSQRT_DIM);       // r_hat = q*deq

  // ---- quantize + dequantize in place (pack/unpack is identity on q) ----
#pragma unroll
  for (int t = 0; t < 8; ++t) {
    v4f v = *reinterpret_cast<v4f*>(&smem[rbase + 4 * t]);
#pragma unroll
    for (int e = 0; e < 4; ++e) {
      const float q = fminf(7.f, fmaxf(-8.f, rintf(v[e] * qmul)));  // RNE == jnp.round
      v[e] = q * deq;
    }
    *reinterpret_cast<v4f*>(&smem[rbase + 4 * t]) = v;
  }

  // ---- inverse transform: s .* (1/sqrt(128)) * H128 * r_hat ----
  tq_hadamard128(smem, wb, L, A1, A2, onesB, dmulI);

  // ---- store: LDS -> global (pure copy; output sign already applied) ----
#if TQ_ASYNC
  asm volatile("s_wait_dscnt 0" ::: "memory");     // stage-2 stores visible in LDS
#pragma unroll
  for (int i = 0; i < 8; ++i) {
    float* gp = reinterpret_cast<float*>(gout + i * 32 + L);
    float* lp = &smem[wb + i * TQ_STRIDE + 4 * L];
    __builtin_amdgcn_global_store_async_from_lds_b128(
        (tq_g_v4i*)(tq_v4i*)gp, (tq_l_v4i*)(tq_v4i*)lp, 0, 0);
  }
  // s_endpgm performs an implicit wait-idle for outstanding async stores
#else
#pragma unroll
  for (int i = 0; i < 8; ++i) {
    const v4f w = *reinterpret_cast<const v4f*>(&smem[wb + i * TQ_STRIDE + 4 * L]);
    float4 v;
    v.x = w[0]; v.y = w[1]; v.z = w[2]; v.w = w[3];
    gout[i * 32 + L] = v;
  }
#endif
}

extern "C" void kernel_launch(void* const* d_in, const int* in_sizes, int n_in,
                              void* d_out, int out_size, void* d_ws, size_t ws_size,
                              hipStream_t stream) {
  const float* x     = (const float*)d_in[0];
  const float* signs = (const float*)d_in[1];
  float*       out   = (float*)d_out;

  const long long totalRows = (long long)in_sizes[0] / TQ_DIM;   // 524288 for the ref shape
  const int blocks = (int)((totalRows + TQ_ROWS_PER_BLOCK - 1) / TQ_ROWS_PER_BLOCK);

  tq_kernel<<<blocks, 256, 0, stream>>>(x, signs, out, totalRows);
}